// HebbianLayer_60060822667614
// MI455X (gfx1250) — compile-verified
//
#include <hip/hip_runtime.h>
#include <hip/hip_bf16.h>
#include <math.h>

#if defined(__HIP_DEVICE_COMPILE__) && \
    !__has_builtin(__builtin_amdgcn_wmma_f32_16x16x4_f32)
#error "missing __builtin_amdgcn_wmma_f32_16x16x4_f32"
#endif

typedef float v2f __attribute__((ext_vector_type(2)));
typedef float v8f __attribute__((ext_vector_type(8)));

#define BSZ   65536
#define DSZ   512
#define LDA   516   // 16-row A panel stride (4 mod 64 -> conflict-free)
#define LDC   516   // C LDS stride for LN path
#define LDCN  772   // C LDS stride for the 768-col nt pass (4 mod 64)

// D = A(16x4) * B(4x16) + C per the CDNA5 VGPR layout tables:
//   A: lanes 0-15 hold (M=lane, K=0/1 in v0/v1), lanes 16-31 hold (M=lane-16, K=2/3)
//   B: v0 = rows K=0 (lanes 0-15) / K=2 (lanes 16-31); v1 = rows K=1 / K=3
//   C: vr = (M=r, N=lane) for lanes 0-15, (M=r+8, N=lane-16) for lanes 16-31
__device__ __forceinline__ v8f wmma4(v2f a, v2f b, v8f c) {
  return __builtin_amdgcn_wmma_f32_16x16x4_f32(false, a, false, b, (short)0, c,
                                               false, false);
}

__device__ __forceinline__ float wred_sum(float v) {
#pragma unroll
  for (int m = 16; m > 0; m >>= 1) v += __shfl_xor(v, m, 32);
  return v;
}
__device__ __forceinline__ float wred_max(float v) {
#pragma unroll
  for (int m = 16; m > 0; m >>= 1) v = fmaxf(v, __shfl_xor(v, m, 32));
  return v;
}
__device__ __forceinline__ float sigmoidf(float x) {
  return 1.0f / (1.0f + expf(-x));
}

__global__ void zero_kernel(float* __restrict__ p, size_t n) {
  size_t i = (size_t)blockIdx.x * blockDim.x + threadIdx.x;
  size_t st = (size_t)gridDim.x * blockDim.x;
  for (; i < n; i += st) p[i] = 0.0f;
}

// ---------------------------------------------------------------------------
// Fused GEMM (NC=512): out[16 rows x 512] = epilogue(A[16x512] @ Wm[512x512])
// Block = 256 thr (8 waves). A panel staged in LDS; each wave owns 4 n-tiles.
// ---------------------------------------------------------------------------
template <bool ADD_REC, bool BIAS, bool RELU, bool DO_LN>
__global__ __launch_bounds__(256) void gemm512_kernel(
    const float* __restrict__ A, const float* __restrict__ Wm,
    const float* __restrict__ rec, const float* __restrict__ bias,
    const float* __restrict__ lng, const float* __restrict__ lnb,
    float* __restrict__ out) {
  __shared__ float As[16 * LDA];
  __shared__ float Cs[DO_LN ? 16 * LDC : 1];
  const int rowBase = blockIdx.x * 16;

  for (int i = threadIdx.x; i < 16 * 512; i += 256) {
    int r = i >> 9, c = i & 511;
    As[r * LDA + c] = A[(size_t)(rowBase + r) * DSZ + c];
  }
  __syncthreads();

  const int wave = threadIdx.x >> 5, lane = threadIdx.x & 31;
  const int half = lane >> 4, l16 = lane & 15;

  for (int nt = wave; nt < 32; nt += 8) {
    const int n0 = nt * 16;
    v8f c = {};
#pragma unroll 4
    for (int k0 = 0; k0 < 512; k0 += 4) {
      const int ka = k0 + 2 * half;
      v2f a, b;
      a.x = As[l16 * LDA + ka];
      a.y = As[l16 * LDA + ka + 1];
      b.x = Wm[(size_t)ka * DSZ + n0 + l16];
      b.y = Wm[(size_t)(ka + 1) * DSZ + n0 + l16];
      c = wmma4(a, b, c);
    }
#pragma unroll
    for (int r = 0; r < 8; ++r) {
      const int m = r + half * 8;
      float v = c[r];
      if (ADD_REC) v += rec[(size_t)(rowBase + m) * DSZ + n0 + l16];
      if (BIAS) v += bias[n0 + l16];
      if (RELU) v = fmaxf(v, 0.0f);
      if (DO_LN)
        Cs[m * LDC + n0 + l16] = v;
      else
        out[(size_t)(rowBase + m) * DSZ + n0 + l16] = v;
    }
  }

  if (DO_LN) {
    __syncthreads();
    for (int r = wave; r < 16; r += 8) {  // 2 rows per wave
      float s = 0.0f, s2 = 0.0f;
#pragma unroll
      for (int i = 0; i < 16; ++i) {
        float v = Cs[r * LDC + lane + i * 32];
        s += v;
        s2 += v * v;
      }
      s = wred_sum(s);
      s2 = wred_sum(s2);
      const float mu = s * (1.0f / 512.0f);
      const float var = s2 * (1.0f / 512.0f) - mu * mu;
      const float inv = rsqrtf(var + 1e-5f);
#pragma unroll
      for (int i = 0; i < 16; ++i) {
        const int cc = lane + i * 32;
        out[(size_t)(rowBase + r) * DSZ + cc] =
            (Cs[r * LDC + cc] - mu) * inv * lng[cc] + lnb[cc];
      }
    }
  }
}

// ---------------------------------------------------------------------------
// nt = h @ W2 + b2  ([16 x 1536] per block, two 768-col passes) fused with the
// neuromodulator epilogue. Writes new_dop/new_ser/new_gab, atomically
// accumulates per-column gaba sums.
// ---------------------------------------------------------------------------
__global__ __launch_bounds__(256) void gemm_nt_kernel(
    const float* __restrict__ h, const float* __restrict__ W2,
    const float* __restrict__ b2, const float* __restrict__ dop_in,
    const float* __restrict__ ser_in, const float* __restrict__ gab_in,
    float* __restrict__ o_dop, float* __restrict__ o_ser,
    float* __restrict__ o_gab, float* __restrict__ gsum) {
  extern __shared__ float smem[];
  float* As = smem;              // 16 * LDA
  float* Cs = smem + 16 * LDA;   // 16 * LDCN
  const int rowBase = blockIdx.x * 16;

  for (int i = threadIdx.x; i < 16 * 512; i += 256) {
    int r = i >> 9, c = i & 511;
    As[r * LDA + c] = h[(size_t)(rowBase + r) * DSZ + c];
  }
  __syncthreads();

  const int wave = threadIdx.x >> 5, lane = threadIdx.x & 31;
  const int half = lane >> 4, l16 = lane & 15;

  for (int pass = 0; pass < 2; ++pass) {
    for (int nt = wave; nt < 48; nt += 8) {
      const int n0 = pass * 768 + nt * 16;  // global column
      const int nl = nt * 16;               // column within this pass
      v8f c = {};
#pragma unroll 4
      for (int k0 = 0; k0 < 512; k0 += 4) {
        const int ka = k0 + 2 * half;
        v2f a, b;
        a.x = As[l16 * LDA + ka];
        a.y = As[l16 * LDA + ka + 1];
        b.x = W2[(size_t)ka * 1536 + n0 + l16];
        b.y = W2[(size_t)(ka + 1) * 1536 + n0 + l16];
        c = wmma4(a, b, c);
      }
      const float bb = b2[n0 + l16];
#pragma unroll
      for (int r = 0; r < 8; ++r) {
        const int m = r + half * 8;
        Cs[m * LDCN + nl + l16] = c[r] + bb;
      }
    }
    __syncthreads();
    // epilogue: thread t owns feature d = pass*256 + t (local cols 3t..3t+2)
    {
      const int t = threadIdx.x;
      const int d = pass * 256 + t;
      float gs = 0.0f;
#pragma unroll 4
      for (int r = 0; r < 16; ++r) {
        const size_t idx = (size_t)(rowBase + r) * DSZ + d;
        const float pd = Cs[r * LDCN + 3 * t + 0];
        const float ps = Cs[r * LDCN + 3 * t + 1];
        const float pg = Cs[r * LDCN + 3 * t + 2];
        const float invs = 1.0f / fmaxf(ps, 1e-6f);
        const float nd = tanhf(dop_in[idx] + pd * invs);
        const float ns = sigmoidf(ser_in[idx] + ps);
        const float ng = sigmoidf(gab_in[idx] + pg * invs);
        o_dop[idx] = nd;
        o_ser[idx] = ns;
        o_gab[idx] = ng;
        gs += ng;
      }
      atomicAdd(&gsum[d], gs);
    }
    __syncthreads();
  }
}

// Per-row softmax stats (max, sum(exp)) — one wave per row.
__global__ __launch_bounds__(256) void rowstats_kernel(
    const float* __restrict__ src, float* __restrict__ mx,
    float* __restrict__ sm) {
  const int wave = threadIdx.x >> 5, lane = threadIdx.x & 31;
  const size_t row = (size_t)blockIdx.x * 8 + wave;
  const float* p = src + row * DSZ;
  float m = -3.0e38f;
#pragma unroll
  for (int i = 0; i < 16; ++i) m = fmaxf(m, p[lane + i * 32]);
  m = wred_max(m);
  float s = 0.0f;
#pragma unroll
  for (int i = 0; i < 16; ++i) s += expf(p[lane + i * 32] - m);
  s = wred_sum(s);
  if (lane == 0) {
    mx[row] = m;
    sm[row] = s;
  }
}

// acc[i,j] += sum_b softmax(src)[b,i] * (alpha[j]*dop[b,j]*ser[b,j])
// Split-K over batch; block = 8 waves covering a 64(i) x 32(j) region.
__global__ __launch_bounds__(256) void einsum_kernel(
    const float* __restrict__ src, const float* __restrict__ mx,
    const float* __restrict__ sm, const float* __restrict__ dop,
    const float* __restrict__ ser, const float* __restrict__ alpha,
    float* __restrict__ acc, int kspan) {
  const int wave = threadIdx.x >> 5, lane = threadIdx.x & 31;
  const int wi = wave & 3, wj = wave >> 2;
  const int i0 = blockIdx.x * 64 + wi * 16;
  const int j0 = blockIdx.y * 32 + wj * 16;
  const int half = lane >> 4, l16 = lane & 15;
  const size_t kb = (size_t)blockIdx.z * kspan;
  const int ia = i0 + l16;
  const int ja = j0 + l16;
  const float al = alpha[ja];

  v8f c = {};
  for (int kk = 0; kk < kspan; kk += 4) {
    const size_t b0 = kb + kk + 2 * half;
    const size_t b1 = b0 + 1;
    const float rs0 = 1.0f / sm[b0];
    const float rs1 = 1.0f / sm[b1];
    v2f a, b;
    a.x = expf(src[b0 * DSZ + ia] - mx[b0]) * rs0;
    a.y = expf(src[b1 * DSZ + ia] - mx[b1]) * rs1;
    b.x = al * dop[b0 * DSZ + ja] * ser[b0 * DSZ + ja];
    b.y = al * dop[b1 * DSZ + ja] * ser[b1 * DSZ + ja];
    c = wmma4(a, b, c);
  }
#pragma unroll
  for (int r = 0; r < 8; ++r) {
    const int ii = i0 + r + half * 8;
    atomicAdd(&acc[(size_t)ii * DSZ + j0 + l16], c[r]);
  }
}

// new_W row = W + softmax(W,row)*acc/B - decay[i]*sigmoid(gsum[i]/B)*W
__global__ __launch_bounds__(256) void wupdate_kernel(
    const float* __restrict__ Wm, const float* __restrict__ acc,
    const float* __restrict__ decay, const float* __restrict__ gsum,
    float* __restrict__ out) {
  __shared__ float red[8];
  const int row = blockIdx.x;
  const int wave = threadIdx.x >> 5, lane = threadIdx.x & 31;
  const float* p = Wm + (size_t)row * DSZ;
  const float v0 = p[threadIdx.x];
  const float v1 = p[threadIdx.x + 256];
  float m = wred_max(fmaxf(v0, v1));
  if (lane == 0) red[wave] = m;
  __syncthreads();
  float mall = red[0];
#pragma unroll
  for (int i = 1; i < 8; ++i) mall = fmaxf(mall, red[i]);
  const float e0 = expf(v0 - mall), e1 = expf(v1 - mall);
  float s = wred_sum(e0 + e1);
  __syncthreads();
  if (lane == 0) red[wave] = s;
  __syncthreads();
  float sall = 0.0f;
#pragma unroll
  for (int i = 0; i < 8; ++i) sall += red[i];
  const float rs = 1.0f / sall;
  const float sd = decay[row] * sigmoidf(gsum[row] * (1.0f / (float)BSZ));
  const float a0 = acc[(size_t)row * DSZ + threadIdx.x] * (1.0f / (float)BSZ);
  const float a1 =
      acc[(size_t)row * DSZ + threadIdx.x + 256] * (1.0f / (float)BSZ);
  out[(size_t)row * DSZ + threadIdx.x] = v0 + e0 * rs * a0 - sd * v0;
  out[(size_t)row * DSZ + threadIdx.x + 256] = v1 + e1 * rs * a1 - sd * v1;
}

extern "C" void kernel_launch(void* const* d_in, const int* in_sizes, int n_in,
                              void* d_out, int out_size, void* d_ws,
                              size_t ws_size, hipStream_t stream) {
  constexpr size_t BD = (size_t)BSZ * DSZ;
  constexpr size_t DD = (size_t)DSZ * DSZ;

  const float* x       = (const float*)d_in[0];
  const float* W       = (const float*)d_in[1];
  const float* Wr      = (const float*)d_in[2];
  const float* alpha   = (const float*)d_in[3];
  const float* decay   = (const float*)d_in[4];
  const float* ln_r_g  = (const float*)d_in[5];
  const float* ln_r_b  = (const float*)d_in[6];
  const float* ln_a_g  = (const float*)d_in[7];
  const float* ln_a_b  = (const float*)d_in[8];
  const float* W1      = (const float*)d_in[9];
  const float* b1      = (const float*)d_in[10];
  const float* W2      = (const float*)d_in[11];
  const float* b2      = (const float*)d_in[12];
  const float* dop_in  = (const float*)d_in[13];
  const float* ser_in  = (const float*)d_in[14];
  const float* gab_in  = (const float*)d_in[15];
  const float* prev    = (const float*)d_in[16];

  float* out   = (float*)d_out;
  float* o_act = out;
  float* o_W   = out + BD;
  float* o_Wr  = o_W + DD;
  float* o_dop = o_Wr + DD;
  float* o_ser = o_dop + BD;
  float* o_gab = o_ser + BD;

  float* ws      = (float*)d_ws;
  float* ws_rec  = ws;               // [B,D]
  float* ws_h    = ws + BD;          // [B,D]
  float* ws_amax = ws + 2 * BD;      // [B]
  float* ws_asum = ws_amax + BSZ;    // [B]
  float* ws_rmax = ws_asum + BSZ;    // [B]
  float* ws_rsum = ws_rmax + BSZ;    // [B]
  float* ws_accA = ws_rsum + BSZ;    // [D,D]
  float* ws_accR = ws_accA + DD;     // [D,D]
  float* ws_gsum = ws_accR + DD;     // [D]

  const dim3 blk(256);
  const dim3 gGemm(BSZ / 16);

  // zero split-K accumulators + gaba column sums (contiguous)
  zero_kernel<<<256, blk, 0, stream>>>(ws_accA, 2 * DD + DSZ);

  // rec_in = LN(prev @ Wr)
  gemm512_kernel<false, false, false, true><<<gGemm, blk, 0, stream>>>(
      prev, Wr, nullptr, nullptr, ln_r_g, ln_r_b, ws_rec);
  // act = LN(relu(x @ W + rec_in))
  gemm512_kernel<true, false, true, true><<<gGemm, blk, 0, stream>>>(
      x, W, ws_rec, nullptr, ln_a_g, ln_a_b, o_act);
  // h = relu(act @ W1 + b1)
  gemm512_kernel<false, true, true, false><<<gGemm, blk, 0, stream>>>(
      o_act, W1, nullptr, b1, nullptr, nullptr, ws_h);

  // nt = h @ W2 + b2, fused neuromodulator epilogue
  const int smem_nt = (16 * LDA + 16 * LDCN) * (int)sizeof(float);
  (void)hipFuncSetAttribute((const void*)gemm_nt_kernel,
                            hipFuncAttributeMaxDynamicSharedMemorySize,
                            smem_nt);
  gemm_nt_kernel<<<gGemm, blk, smem_nt, stream>>>(
      ws_h, W2, b2, dop_in, ser_in, gab_in, o_dop, o_ser, o_gab, ws_gsum);

  // row softmax stats
  rowstats_kernel<<<BSZ / 8, blk, 0, stream>>>(o_act, ws_amax, ws_asum);
  rowstats_kernel<<<BSZ / 8, blk, 0, stream>>>(ws_rec, ws_rmax, ws_rsum);

  // split-K einsums: softmax(src)^T @ (alpha*dop*ser)
  const int SPLITS = 16;
  const int kspan = BSZ / SPLITS;
  const dim3 gE(DSZ / 64, DSZ / 32, SPLITS);
  einsum_kernel<<<gE, blk, 0, stream>>>(o_act, ws_amax, ws_asum, o_dop, o_ser,
                                        alpha, ws_accA, kspan);
  einsum_kernel<<<gE, blk, 0, stream>>>(ws_rec, ws_rmax, ws_rsum, o_dop, o_ser,
                                        alpha, ws_accR, kspan);

  // final weight updates
  wupdate_kernel<<<DSZ, blk, 0, stream>>>(W, ws_accA, decay, ws_gsum, o_W);
  wupdate_kernel<<<DSZ, blk, 0, stream>>>(Wr, ws_accR, decay, ws_gsum, o_Wr);
}